// PICAN_51436528337471
// MI455X (gfx1250) — compile-verified
//
#include <hip/hip_runtime.h>
#include <hip/hip_bf16.h>
#include <math.h>

// GGNN: L layers of  m = h@W ; agg = scatter_add(edge_attr * m[src], dst) ;
//       h = GRUCell(agg, h) ; then softmax(h@w_ro + b_ro).
// fp32 WMMA (V_WMMA_F32_16X16X4_F32) for all GEMMs: workload is bound by the
// edge scatter (~6.5 GB @ 23.3 TB/s), so full-precision tensor ops are free.
// LDS staging of the 16-row strips uses GLOBAL_LOAD_ASYNC_TO_LDS_B128
// (ASYNCcnt path) instead of a VGPR round-trip.

typedef __attribute__((ext_vector_type(2))) float v2f;
typedef __attribute__((ext_vector_type(8))) float v8f;

#define DD 256          // hidden dim
#define KOUT 32         // readout classes (== wave32 lanes)
#define SROW 260        // LDS row stride (260%64=4 -> conflict-free col reads,
                        // 260*4B row pitch keeps 16B alignment for b128 ops)

__device__ __forceinline__ v8f wmma_f32(v2f a, v2f b, v8f c) {
  // 8 args: (neg_a, A, neg_b, B, c_mod, C, reuse_a, reuse_b)
  return __builtin_amdgcn_wmma_f32_16x16x4_f32(false, a, false, b, (short)0, c,
                                               false, false);
}

// async 16B global -> LDS copy (no VGPR round-trip), tracked by ASYNCcnt
__device__ __forceinline__ void async_cp16(void* lds, const void* gaddr) {
  unsigned int l = (unsigned int)(size_t)lds;   // LDS aperture uses addr[31:0]
  asm volatile("global_load_async_to_lds_b128 %0, %1, off"
               :: "v"(l), "v"(gaddr) : "memory");
}
__device__ __forceinline__ void async_wait0() {
  asm volatile("s_wait_asynccnt 0" ::: "memory");
}

// ---------------------------------------------------------------- init / zero
__global__ void __launch_bounds__(256) ggnn_init_h(const float* __restrict__ x,
                                                   float* __restrict__ h, int n) {
  long long t = (long long)blockIdx.x * blockDim.x + threadIdx.x;
  if (t >= (long long)n * DD) return;
  int c = (int)(t & (DD - 1));
  long long node = t >> 8;
  h[t] = (c == 0) ? x[node] : 0.0f;
}

__global__ void __launch_bounds__(256) ggnn_zero(float4* __restrict__ p, long long n4) {
  long long t = (long long)blockIdx.x * blockDim.x + threadIdx.x;
  if (t < n4) p[t] = make_float4(0.f, 0.f, 0.f, 0.f);
}

// --------------------------------------------------------- m = h @ W  (WMMA)
// One block per 16-row strip. h strip async-staged into LDS, 8 waves each
// compute two 16x16 output tiles, K-loop in steps of 4 (fp32 WMMA).
__global__ void __launch_bounds__(256) ggnn_gemm_m(const float* __restrict__ h,
                                                   const float* __restrict__ W,
                                                   float* __restrict__ m) {
  __shared__ float sA[16 * SROW];
  const long long row0 = (long long)blockIdx.x * 16;

  // stage 16x256 strip of h -> LDS via async b128 copies
  const float4* src4 = (const float4*)(h + row0 * DD);
  for (int i = threadIdx.x; i < 16 * DD / 4; i += blockDim.x) {
    int r = i >> 6, cg = i & 63;
    async_cp16(&sA[r * SROW + cg * 4], &src4[i]);
  }
  async_wait0();
  __syncthreads();

  const int lane = threadIdx.x & 31, wave = threadIdx.x >> 5;
  const int l16 = lane & 15;
  const int koff = (lane >> 4) << 1;     // 0 or 2 (A/B half-wave K offset)

  for (int ct = 0; ct < 2; ++ct) {
    const int col0 = (wave * 2 + ct) * 16;
    v8f acc = {};
#pragma unroll 4
    for (int kk = 0; kk < DD; kk += 4) {
      v2f a, b;
      a.x = sA[l16 * SROW + kk + koff];
      a.y = sA[l16 * SROW + kk + koff + 1];
      const float* wp = W + (size_t)(kk + koff) * DD + col0 + l16;
      b.x = wp[0];
      b.y = wp[DD];
      acc = wmma_f32(a, b, acc);
    }
    const int rsel = (lane >> 4) << 3;   // rows r / r+8
    float* out = m + (row0 + rsel) * DD + col0 + l16;
#pragma unroll
    for (int r = 0; r < 8; ++r) out[(size_t)r * DD] = acc[r];
  }
}

// ------------------------------------------- agg[dst] += edge_attr * m[src]
// 64 threads per edge, float4 gather, 4 global_atomic_add_f32 per thread.
__global__ void __launch_bounds__(256) ggnn_scatter(const float* __restrict__ m,
                                                    const long long* __restrict__ ei,
                                                    const float* __restrict__ ea,
                                                    float* __restrict__ agg, int E) {
  long long t = (long long)blockIdx.x * blockDim.x + threadIdx.x;
  long long e = t >> 6;                   // DD/4 = 64 groups per edge
  if (e >= E) return;
  int g = (int)(t & 63) * 4;
  size_t s = (size_t)ei[e];               // edge_index[0][e]
  size_t d = (size_t)ei[(long long)E + e];// edge_index[1][e]
  float w = ea[e];
  float4 mv = *(const float4*)(m + s * DD + g);
  float* ap = agg + d * DD + g;
  atomicAdd(ap + 0, w * mv.x);
  atomicAdd(ap + 1, w * mv.y);
  atomicAdd(ap + 2, w * mv.z);
  atomicAdd(ap + 3, w * mv.w);
}

// ------------------------------------------------------- fused GRU cell step
// Per 16-row strip: gi = agg@w_ih^T + b_ih, gh = h@w_hh^T + b_hh (6 WMMA
// accumulators per wave per column tile: i_r,i_z,i_n,h_r,h_z,h_n), then the
// gate nonlinearity, writing h_next.
__global__ void __launch_bounds__(256) ggnn_gru(const float* __restrict__ agg,
                                                const float* __restrict__ h,
                                                const float* __restrict__ w_ih,
                                                const float* __restrict__ w_hh,
                                                const float* __restrict__ b_ih,
                                                const float* __restrict__ b_hh,
                                                float* __restrict__ h_next) {
  __shared__ float sA[16 * SROW];
  __shared__ float sH[16 * SROW];
  const long long row0 = (long long)blockIdx.x * 16;

  const float4* a4 = (const float4*)(agg + row0 * DD);
  const float4* h4 = (const float4*)(h + row0 * DD);
  for (int i = threadIdx.x; i < 16 * DD / 4; i += blockDim.x) {
    int r = i >> 6, cg = i & 63;
    async_cp16(&sA[r * SROW + cg * 4], &a4[i]);
    async_cp16(&sH[r * SROW + cg * 4], &h4[i]);
  }
  async_wait0();
  __syncthreads();

  const int lane = threadIdx.x & 31, wave = threadIdx.x >> 5;
  const int l16 = lane & 15;
  const int koff = (lane >> 4) << 1;

  for (int ct = 0; ct < 2; ++ct) {
    const int col0 = (wave * 2 + ct) * 16;
    const int jr = 0 * DD + col0 + l16;   // torch gate order [r|z|n]
    const int jz = 1 * DD + col0 + l16;
    const int jn = 2 * DD + col0 + l16;

    v8f cir = {}, ciz = {}, cin = {}, chr_ = {}, chz = {}, chn = {};
#pragma unroll 2
    for (int kk = 0; kk < DD; kk += 4) {
      v2f aA, aH;
      aA.x = sA[l16 * SROW + kk + koff];
      aA.y = sA[l16 * SROW + kk + koff + 1];
      aH.x = sH[l16 * SROW + kk + koff];
      aH.y = sH[l16 * SROW + kk + koff + 1];
      // B[k][j] = w[j*D + k] : contiguous float2 per lane (8B aligned)
      v2f bir = *(const v2f*)(w_ih + (size_t)jr * DD + kk + koff);
      v2f biz = *(const v2f*)(w_ih + (size_t)jz * DD + kk + koff);
      v2f bin = *(const v2f*)(w_ih + (size_t)jn * DD + kk + koff);
      v2f bhr = *(const v2f*)(w_hh + (size_t)jr * DD + kk + koff);
      v2f bhz = *(const v2f*)(w_hh + (size_t)jz * DD + kk + koff);
      v2f bhn = *(const v2f*)(w_hh + (size_t)jn * DD + kk + koff);
      cir  = wmma_f32(aA, bir, cir);
      ciz  = wmma_f32(aA, biz, ciz);
      cin  = wmma_f32(aA, bin, cin);
      chr_ = wmma_f32(aH, bhr, chr_);
      chz  = wmma_f32(aH, bhz, chz);
      chn  = wmma_f32(aH, bhn, chn);
    }

    const float bir_s = b_ih[jr], biz_s = b_ih[jz], bin_s = b_ih[jn];
    const float bhr_s = b_hh[jr], bhz_s = b_hh[jz], bhn_s = b_hh[jn];
    const int rsel = (lane >> 4) << 3;
    float* out = h_next + (row0 + rsel) * DD + col0 + l16;
#pragma unroll
    for (int r = 0; r < 8; ++r) {
      float ir = cir[r] + bir_s, iz = ciz[r] + biz_s, in_ = cin[r] + bin_s;
      float hr = chr_[r] + bhr_s, hz = chz[r] + bhz_s, hn = chn[r] + bhn_s;
      float rg = 1.0f / (1.0f + __expf(-(ir + hr)));
      float zg = 1.0f / (1.0f + __expf(-(iz + hz)));
      float ng = tanhf(in_ + rg * hn);
      float hp = sH[(rsel + r) * SROW + col0 + l16];
      out[(size_t)r * DD] = (1.0f - zg) * ng + zg * hp;
    }
  }
}

// ------------------------------------------- readout + softmax (wave32/node)
__global__ void __launch_bounds__(256) ggnn_readout(const float* __restrict__ h,
                                                    const float* __restrict__ w_ro,
                                                    const float* __restrict__ b_ro,
                                                    float* __restrict__ out, int n) {
  int node = blockIdx.x * 8 + (threadIdx.x >> 5);
  int k = threadIdx.x & 31;              // lane == class (KOUT == 32)
  if (node >= n) return;
  const float* hp = h + (size_t)node * DD;
  float acc = b_ro[k];
#pragma unroll 8
  for (int j = 0; j < DD; ++j) acc += hp[j] * w_ro[j * KOUT + k];
  float mx = acc;
  for (int off = 16; off; off >>= 1) mx = fmaxf(mx, __shfl_xor(mx, off, 32));
  float ex = __expf(acc - mx);
  float sm = ex;
  for (int off = 16; off; off >>= 1) sm += __shfl_xor(sm, off, 32);
  out[(size_t)node * KOUT + k] = ex / sm;
}

// ---------------------------------------------------------------------------
extern "C" void kernel_launch(void* const* d_in, const int* in_sizes, int n_in,
                              void* d_out, int out_size, void* d_ws, size_t ws_size,
                              hipStream_t stream) {
  const float*     x      = (const float*)d_in[0];
  const long long* ei     = (const long long*)d_in[1];   // int64 (2,E)
  const float*     eattr  = (const float*)d_in[2];
  const float*     weight = (const float*)d_in[3];       // (L,D,D)
  const float*     w_ih   = (const float*)d_in[4];       // (3D,D)
  const float*     w_hh   = (const float*)d_in[5];
  const float*     b_ih   = (const float*)d_in[6];
  const float*     b_hh   = (const float*)d_in[7];
  const float*     w_ro   = (const float*)d_in[8];       // (D,K)
  const float*     b_ro   = (const float*)d_in[9];
  float* out = (float*)d_out;

  const int N = in_sizes[0];                 // x is (N,1)
  const int E = in_sizes[2];                 // edge_attr is (E,)
  const int L = in_sizes[3] / (DD * DD);     // (L,D,D)

  const size_t nd = (size_t)N * DD;
  float* h0  = (float*)d_ws;                 // N*D
  float* h1  = h0 + nd;                      // N*D
  float* m   = h1 + nd;                      // N*D
  float* agg = m + nd;                       // N*D

  const int strips = (N + 15) / 16;          // N=50000 -> 3125 exact strips
  const long long nd4 = (long long)nd / 4;

  ggnn_init_h<<<(int)((nd + 255) / 256), 256, 0, stream>>>(x, h0, N);

  float* hc = h0;
  float* hn = h1;
  for (int l = 0; l < L; ++l) {
    ggnn_gemm_m<<<strips, 256, 0, stream>>>(hc, weight + (size_t)l * DD * DD, m);
    ggnn_zero<<<(int)((nd4 + 255) / 256), 256, 0, stream>>>((float4*)agg, nd4);
    long long sthreads = (long long)E * 64;
    ggnn_scatter<<<(int)((sthreads + 255) / 256), 256, 0, stream>>>(m, ei, eattr, agg, E);
    ggnn_gru<<<strips, 256, 0, stream>>>(agg, hc, w_ih, w_hh, b_ih, b_hh, hn);
    float* t = hc; hc = hn; hn = t;
  }

  ggnn_readout<<<(N + 7) / 8, 256, 0, stream>>>(hc, w_ro, b_ro, out, N);
}